// EfficientNonSamplingFactorizationMachines_33741263078160
// MI455X (gfx1250) — compile-verified
//
#include <hip/hip_runtime.h>
#include <hip/hip_bf16.h>

typedef __attribute__((ext_vector_type(2))) float v2f;
typedef __attribute__((ext_vector_type(8))) float v8f;

#define FEAT 20
#define DIM 64
#define NEG_W 0.5f
#define GR_ROWS 16   // K-rows staged in LDS per outer iteration

// ---------------------------------------------------------------------------
// Zero the Gram accumulators and the output scalar.
// ---------------------------------------------------------------------------
__global__ void zero_kernel(float* __restrict__ Ag, float* __restrict__ Bg,
                            float* __restrict__ out) {
    int i = blockIdx.x * blockDim.x + threadIdx.x;
    if (i < DIM * DIM) { Ag[i] = 0.0f; Bg[i] = 0.0f; }
    if (i == 0) out[0] = 0.0f;
}

// ---------------------------------------------------------------------------
// Stage 1: per-entity FM terms.
// One block (64 threads = 2 waves) per entity row; thread d owns dimension d.
// Coalesced: 64 lanes read one contiguous 256B embedding row per feature.
// ---------------------------------------------------------------------------
__global__ void entity_kernel(const float* __restrict__ emb,
                              const float* __restrict__ w,
                              const int*   __restrict__ fidx,
                              const float* __restrict__ h1,
                              const float* __restrict__ bias, int use_bias,
                              float* __restrict__ P,   // [N, 64] per-dim sums
                              float* __restrict__ S)   // [N] scalar term
{
    const int n = blockIdx.x;
    const int d = threadIdx.x;      // 0..63

    __shared__ int   fi[FEAT];
    __shared__ float red[DIM];

    if (d < FEAT) fi[d] = fidx[(size_t)n * FEAT + d];
    __syncthreads();

    float s = 0.0f, sq = 0.0f;
#pragma unroll
    for (int f = 0; f < FEAT; ++f) {
        float g = emb[(size_t)fi[f] * DIM + d];
        s  += g;
        sq += g * g;
    }
    P[(size_t)n * DIM + d] = s;

    red[d] = 0.5f * (s * s - sq) * h1[d];
    __syncthreads();
#pragma unroll
    for (int off = 32; off > 0; off >>= 1) {
        if (d < off) red[d] += red[d + off];
        __syncthreads();
    }
    if (d == 0) {
        float wsum = 0.0f;
#pragma unroll
        for (int f = 0; f < FEAT; ++f) wsum += w[fi[f]];
        S[n] = red[0] + wsum + (use_bias ? bias[0] : 0.0f);
    }
}

// ---------------------------------------------------------------------------
// Stage 2: G += P^T P  (64x64 Gram of an [N,64] matrix) via
// V_WMMA_F32_16X16X4_F32.
//
// 4 waves per block; wave w owns output row-strip [16w, 16w+16) -> 4 tiles
// of 8 accumulator VGPRs (32 VGPRs/wave -> no spills).  Each outer iteration
// stages GR_ROWS=16 rows of P (4 KB) in LDS shared by all 4 waves, then each
// wave runs 4 K-steps (K=4 each) issuing 4 WMMAs per step = 16 WMMAs per
// wave per barrier pair.  EXEC is all-ones at every WMMA (uniform flow).
// ---------------------------------------------------------------------------
__global__ void gram_kernel(const float* __restrict__ P, int N,
                            float* __restrict__ G)
{
    const int tid  = threadIdx.x;        // 0..127
    const int wave = tid >> 5;           // 0..3 -> row-tile index ti
    const int lane = tid & 31;
    const int half = lane >> 4;          // 0 or 1
    const int lh   = lane & 15;          // 0..15

    __shared__ float tile[GR_ROWS * DIM];   // 16 rows of P

    v8f acc[4];
#pragma unroll
    for (int tj = 0; tj < 4; ++tj)
        acc[tj] = (v8f){0,0,0,0,0,0,0,0};

    for (int k0 = blockIdx.x * GR_ROWS; k0 < N; k0 += gridDim.x * GR_ROWS) {
        // stage 16x64 block into LDS (coalesced; zero-pad any tail rows)
#pragma unroll
        for (int i = 0; i < (GR_ROWS * DIM) / 128; ++i) {
            int e   = tid + 128 * i;          // 0..1023
            int row = k0 + (e >> 6);
            tile[e] = (row < N) ? P[(size_t)row * DIM + (e & 63)] : 0.0f;
        }
        __syncthreads();

#pragma unroll
        for (int ks = 0; ks < GR_ROWS / 4; ++ks) {
            const float* tp = &tile[ks * 4 * DIM];
            // A-frag: 16x4 slice of P^T for this wave's row strip
            //   lanes 0-15: K=0 (VGPR0), K=1 (VGPR1); lanes 16-31: K=2,3
            v2f a;
            a.x = tp[(2 * half + 0) * DIM + wave * 16 + lh];
            a.y = tp[(2 * half + 1) * DIM + wave * 16 + lh];
#pragma unroll
            for (int tj = 0; tj < 4; ++tj) {
                // B-frag: 4x16 slice of P
                //   VGPR0: rows K0/K1 across lane halves; VGPR1: K2/K3
                v2f b;
                b.x = tp[(half + 0) * DIM + tj * 16 + lh];
                b.y = tp[(half + 2) * DIM + tj * 16 + lh];
                acc[tj] = __builtin_amdgcn_wmma_f32_16x16x4_f32(
                    false, a, false, b, (short)0, acc[tj], false, false);
            }
        }
        __syncthreads();
    }

    // C/D layout: VGPR r, lanes 0-15 -> M = r, lanes 16-31 -> M = r + 8
#pragma unroll
    for (int tj = 0; tj < 4; ++tj)
#pragma unroll
        for (int r = 0; r < 8; ++r) {
            int row = wave * 16 + r + 8 * half;
            int col = tj * 16 + lh;
            atomicAdd(&G[row * DIM + col], acc[tj][r]);
        }
}

// ---------------------------------------------------------------------------
// Stage 3: loss_neg = NEG_W * sum_ij h2_i h2_j A_ij B_ij   (single block)
// ---------------------------------------------------------------------------
__global__ void negloss_kernel(const float* __restrict__ A,
                               const float* __restrict__ B,
                               const float* __restrict__ h2,
                               float* __restrict__ out)
{
    const int i = threadIdx.x;           // 0..63
    __shared__ float red[DIM];
    float s = 0.0f;
    for (int j = 0; j < DIM; ++j)
        s += h2[j] * A[i * DIM + j] * B[i * DIM + j];
    red[i] = h2[i] * s;
    __syncthreads();
#pragma unroll
    for (int off = 32; off > 0; off >>= 1) {
        if (i < off) red[i] += red[i + off];
        __syncthreads();
    }
    if (i == 0) atomicAdd(out, NEG_W * red[0]);
}

// ---------------------------------------------------------------------------
// Stage 4: positive-pair loss. One wave32 per pair (grid-stride); lane d
// handles dims d and d+32; wave reduction via shfl_xor; per-wave atomicAdd.
//   y = sum_d h2[d] p[u][d] q[v][d] + su[u] + sv[v]
//   loss += (1-NEG_W) y^2 - 2y
// ---------------------------------------------------------------------------
__global__ void posloss_kernel(const float* __restrict__ P,
                               const float* __restrict__ Su,
                               const float* __restrict__ Q,
                               const float* __restrict__ Sv,
                               const float* __restrict__ h2,
                               const int*   __restrict__ pu,
                               const int*   __restrict__ pv,
                               int npairs, float* __restrict__ out)
{
    const int lane   = threadIdx.x & 31;
    const int wave   = (blockIdx.x * blockDim.x + threadIdx.x) >> 5;
    const int nwaves = (gridDim.x * blockDim.x) >> 5;

    const float h2a = h2[lane];
    const float h2b = h2[lane + 32];

    float acc = 0.0f;
    for (int p = wave; p < npairs; p += nwaves) {
        const int u = pu[p];
        const int v = pv[p];
        const float* pr = P + (size_t)u * DIM;
        const float* qr = Q + (size_t)v * DIM;
        float t = h2a * pr[lane] * qr[lane]
                + h2b * pr[lane + 32] * qr[lane + 32];
#pragma unroll
        for (int off = 16; off > 0; off >>= 1)
            t += __shfl_xor(t, off, 32);
        if (lane == 0) {
            float y = t + Su[u] + Sv[v];
            acc += (1.0f - NEG_W) * y * y - 2.0f * y;
        }
    }
    if (lane == 0) atomicAdd(out, acc);
}

// ---------------------------------------------------------------------------
extern "C" void kernel_launch(void* const* d_in, const int* in_sizes, int n_in,
                              void* d_out, int out_size, void* d_ws, size_t ws_size,
                              hipStream_t stream) {
    const float* user_emb = (const float*)d_in[0];
    const float* item_emb = (const float*)d_in[1];
    const float* w_user   = (const float*)d_in[2];
    const float* w_item   = (const float*)d_in[3];
    const float* bias     = (const float*)d_in[4];
    const float* h1       = (const float*)d_in[5];
    const float* h2       = (const float*)d_in[6];
    const int*   ufeat    = (const int*)d_in[7];
    const int*   ifeat    = (const int*)d_in[8];
    const int*   pu       = (const int*)d_in[9];
    const int*   pv       = (const int*)d_in[10];
    float* out = (float*)d_out;

    const int U  = in_sizes[7] / FEAT;   // 16384
    const int V  = in_sizes[8] / FEAT;   // 50000
    const int Pn = in_sizes[9];          // 65536

    // workspace layout (floats)
    float* ws   = (float*)d_ws;
    float* p_ud = ws;                                  // U*64
    float* q_vd = p_ud + (size_t)U * DIM;              // V*64
    float* su   = q_vd + (size_t)V * DIM;              // U
    float* sv   = su + U;                              // V
    float* Ag   = sv + V;                              // 64*64
    float* Bg   = Ag + DIM * DIM;                      // 64*64

    zero_kernel<<<(DIM * DIM + 255) / 256, 256, 0, stream>>>(Ag, Bg, out);

    entity_kernel<<<U, DIM, 0, stream>>>(user_emb, w_user, ufeat, h1, bias, 1,
                                         p_ud, su);
    entity_kernel<<<V, DIM, 0, stream>>>(item_emb, w_item, ifeat, h1, bias, 0,
                                         q_vd, sv);

    gram_kernel<<<128, 128, 0, stream>>>(p_ud, U, Ag);
    gram_kernel<<<128, 128, 0, stream>>>(q_vd, V, Bg);

    negloss_kernel<<<1, DIM, 0, stream>>>(Ag, Bg, h2, out);

    posloss_kernel<<<256, 256, 0, stream>>>(p_ud, su, q_vd, sv, h2,
                                            pu, pv, Pn, out);
}